// Model_20701742366783
// MI455X (gfx1250) — compile-verified
//
#include <hip/hip_runtime.h>
#include <hip/hip_bf16.h>

typedef __attribute__((ext_vector_type(2))) float v2f;
typedef __attribute__((ext_vector_type(8))) float v8f;

#define B_    16
#define L_    512
#define N_    321
#define MK_   4
#define T_    (N_ + MK_)      // 325
#define D_    512
#define S_    16
#define DTR_  32
#define PRED_ 96
#define MROWS (B_ * T_)       // 5200 -> 325 M-tiles of 16

// ---------------------------------------------------------------------------
// 1) per-(b,n) mean / std over L
// ---------------------------------------------------------------------------
__global__ void stats_kernel(const float* __restrict__ x_enc,
                             float* __restrict__ mean_o,
                             float* __restrict__ std_o) {
    int bn = blockIdx.x;                    // 0 .. B*N-1
    int b = bn / N_, n = bn % N_;
    const float* base = x_enc + (size_t)b * L_ * N_ + n;
    float s = 0.f, s2 = 0.f;
    for (int l = threadIdx.x; l < L_; l += 256) {
        float v = base[(size_t)l * N_];
        s += v; s2 += v * v;
    }
    __shared__ float sh0[256], sh1[256];
    sh0[threadIdx.x] = s; sh1[threadIdx.x] = s2;
    __syncthreads();
    for (int off = 128; off > 0; off >>= 1) {
        if (threadIdx.x < off) {
            sh0[threadIdx.x] += sh0[threadIdx.x + off];
            sh1[threadIdx.x] += sh1[threadIdx.x + off];
        }
        __syncthreads();
    }
    if (threadIdx.x == 0) {
        float m = sh0[0] * (1.f / L_);
        float var = sh1[0] * (1.f / L_) - m * m;
        mean_o[bn] = m;
        std_o[bn]  = sqrtf(var + 1e-5f);
    }
}

// ---------------------------------------------------------------------------
// 2) tok[b,t,l] = normalized transpose of x_enc / x_mark_enc
// ---------------------------------------------------------------------------
__global__ void tok_kernel(const float* __restrict__ x_enc,
                           const float* __restrict__ x_mark,
                           const float* __restrict__ mean_i,
                           const float* __restrict__ std_i,
                           float* __restrict__ tok) {
    size_t idx = (size_t)blockIdx.x * 256 + threadIdx.x;   // B*T*L total
    int l = (int)(idx % L_);
    size_t r = idx / L_;
    int t = (int)(r % T_);
    int b = (int)(r / T_);
    float v;
    if (t < N_) {
        int bn = b * N_ + t;
        v = (x_enc[(size_t)b * L_ * N_ + (size_t)l * N_ + t] - mean_i[bn]) / std_i[bn];
    } else {
        v = x_mark[(size_t)b * L_ * MK_ + (size_t)l * MK_ + (t - N_)];
    }
    tok[idx] = v;
}

// ---------------------------------------------------------------------------
// 3) generic fp32 WMMA GEMM:  C[M,N] = act(A[M,K] @ W[K,N] + bias) + addsrc
//    Each wave: 1 M-tile x NT N-tiles. A-fragment loaded once per K-step as a
//    single b128 (float4) and reused across NT V_WMMA_F32_16X16X4_F32 ops.
//    act: 0=none, 1=sigmoid, 2=softplus
// ---------------------------------------------------------------------------
template <int NT>
__global__ void wmma_gemm_kernel(const float* __restrict__ A, int lda,
                                 const float* __restrict__ W, int K, int N,
                                 const float* __restrict__ bias,
                                 const float* __restrict__ addsrc, int ld_add,
                                 float* __restrict__ C, int ldc,
                                 int Mtiles, int act) {
    int wave = (int)((blockIdx.x * blockDim.x + threadIdx.x) >> 5);
    int lane = threadIdx.x & 31;
    int Ngroups = N / (16 * NT);
    int groups = Mtiles * Ngroups;
    if (wave >= groups) return;           // wave-uniform: EXEC all-ones below

    int mt = wave / Ngroups, ng = wave % Ngroups;
    int m0 = mt << 4, n0 = ng * (16 * NT);
    int l15 = lane & 15;
    int hi  = lane >> 4;                  // 0: K rows 0/1, 1: K rows 2/3
    int koff = hi << 1;

    const float4* Arow4 = (const float4*)(A + (size_t)(m0 + l15) * lda);
    const float* Wcol = W + n0 + l15;

    v8f acc[NT];
#pragma unroll
    for (int j = 0; j < NT; ++j) acc[j] = {};

#pragma unroll 2
    for (int k = 0; k < K; k += 4) {
        float4 av = Arow4[k >> 2];                 // A[m][k..k+3], one b128
        v2f a;
        a[0] = hi ? av.z : av.x;
        a[1] = hi ? av.w : av.y;
        const float* Wk = Wcol + (size_t)(k + koff) * N;
#pragma unroll
        for (int j = 0; j < NT; ++j) {
            v2f bf;
            bf[0] = Wk[j * 16];
            bf[1] = Wk[N + j * 16];
            acc[j] = __builtin_amdgcn_wmma_f32_16x16x4_f32(
                /*neg_a=*/false, a, /*neg_b=*/false, bf,
                /*c_mod=*/(short)0, acc[j], /*reuse_a=*/false, /*reuse_b=*/false);
        }
    }

    int mbase = m0 + (hi << 3);
#pragma unroll
    for (int j = 0; j < NT; ++j) {
        int col = n0 + j * 16 + l15;
        float bval = bias ? bias[col] : 0.f;
#pragma unroll
        for (int r = 0; r < 8; ++r) {
            float v = acc[j][r] + bval;
            if (act == 1)      v = 1.f / (1.f + __expf(-v));
            else if (act == 2) v = (v > 20.f) ? v : log1pf(__expf(v));
            size_t row = (size_t)(mbase + r);
            if (addsrc) v += addsrc[row * ld_add + col];
            C[row * ldc + col] = v;
        }
    }
}

// ---------------------------------------------------------------------------
// 4) selective scan: one thread per (b,d), S=16 state in registers.
//    dir=0 forward (writes ycat[...,0:D]), dir=1 backward (writes ycat[...,D:2D])
// ---------------------------------------------------------------------------
__global__ void scan_kernel(const float* __restrict__ enc,
                            const float* __restrict__ dtbuf,
                            const float* __restrict__ xz,
                            const float* __restrict__ A_log,
                            const float* __restrict__ Dskip,
                            float* __restrict__ ycat,
                            int dir) {
    int d = blockIdx.x * blockDim.x + threadIdx.x;   // grid.x = D/256
    int b = blockIdx.y;

    float Aval[S_];
#pragma unroll
    for (int s = 0; s < S_; ++s) Aval[s] = -__expf(A_log[d * S_ + s]);
    float dskip = Dskip[d];

    float h[S_];
#pragma unroll
    for (int s = 0; s < S_; ++s) h[s] = 0.f;

    int dirOff = dir ? D_ : 0;
    for (int step = 0; step < T_; ++step) {
        int t = dir ? (T_ - 1 - step) : step;
        size_t row = (size_t)b * T_ + t;
        float dt = dtbuf[row * D_ + d];
        float u  = enc[row * D_ + d];
        // prefetch next step's operands (global_prefetch_b8) to overlap the
        // dependent v_exp chain with memory
        if (step + 1 < T_) {
            int t2 = dir ? (t - 1) : (t + 1);
            size_t row2 = (size_t)b * T_ + t2;
            __builtin_prefetch(&dtbuf[row2 * D_ + d], 0, 1);
            __builtin_prefetch(&enc[row2 * D_ + d], 0, 1);
        }
        const float* xzrow = xz + row * 64;          // [0:32)=dt-rank, [32:48)=B, [48:64)=C
        float du = dt * u;
        float acc = 0.f;
#pragma unroll
        for (int s = 0; s < S_; ++s) {
            float Bc = xzrow[DTR_ + s];
            float Cc = xzrow[DTR_ + S_ + s];
            h[s] = __expf(dt * Aval[s]) * h[s] + du * Bc;
            acc += h[s] * Cc;
        }
        ycat[row * (2 * D_) + dirOff + d] = acc + u * dskip;
    }
}

// ---------------------------------------------------------------------------
// 5) fused = z*yf + (1-z)*yb   (z already sigmoid'ed by GEMM epilogue)
// ---------------------------------------------------------------------------
__global__ void fuse_kernel(const float* __restrict__ ycat,
                            float* __restrict__ z_io) {
    size_t idx = (size_t)blockIdx.x * 256 + threadIdx.x;    // B*T*D total
    size_t row = idx / D_;
    int d = (int)(idx % D_);
    float yf = ycat[row * (2 * D_) + d];
    float yb = ycat[row * (2 * D_) + D_ + d];
    float z = z_io[idx];
    z_io[idx] = z * yf + (1.f - z) * yb;
}

// ---------------------------------------------------------------------------
// 6) LayerNorm over D, in place. One block (256 thr, 2 elem/thr) per row.
// ---------------------------------------------------------------------------
__global__ void ln_kernel(float* __restrict__ hbuf,
                          const float* __restrict__ g,
                          const float* __restrict__ beta) {
    int row = blockIdx.x;
    float* hr = hbuf + (size_t)row * D_;
    int tid = threadIdx.x;
    float v0 = hr[tid], v1 = hr[tid + 256];
    __shared__ float sh0[256], sh1[256];
    sh0[tid] = v0 + v1; sh1[tid] = v0 * v0 + v1 * v1;
    __syncthreads();
    for (int off = 128; off > 0; off >>= 1) {
        if (tid < off) { sh0[tid] += sh0[tid + off]; sh1[tid] += sh1[tid + off]; }
        __syncthreads();
    }
    __shared__ float s_mean, s_rstd;
    if (tid == 0) {
        float m = sh0[0] * (1.f / D_);
        float var = sh1[0] * (1.f / D_) - m * m;
        s_mean = m;
        s_rstd = rsqrtf(var + 1e-5f);
    }
    __syncthreads();
    float m = s_mean, rs = s_rstd;
    hr[tid]       = (v0 - m) * rs * g[tid]       + beta[tid];
    hr[tid + 256] = (v1 - m) * rs * g[tid + 256] + beta[tid + 256];
}

// ---------------------------------------------------------------------------
// 7) dec[b,p,n] = proj[b*T+n, p] * std[b,n] + mean[b,n]   (n < N)
// ---------------------------------------------------------------------------
__global__ void out_kernel(const float* __restrict__ proj,
                           const float* __restrict__ mean_i,
                           const float* __restrict__ std_i,
                           float* __restrict__ out) {
    size_t idx = (size_t)blockIdx.x * 256 + threadIdx.x;
    if (idx >= (size_t)B_ * PRED_ * N_) return;
    int n = (int)(idx % N_);
    size_t r = idx / N_;
    int p = (int)(r % PRED_);
    int b = (int)(r / PRED_);
    float v = proj[((size_t)b * T_ + n) * PRED_ + p];
    out[idx] = v * std_i[b * N_ + n] + mean_i[b * N_ + n];
}

// ---------------------------------------------------------------------------
// launch
// ---------------------------------------------------------------------------
extern "C" void kernel_launch(void* const* d_in, const int* in_sizes, int n_in,
                              void* d_out, int out_size, void* d_ws, size_t ws_size,
                              hipStream_t stream) {
    const float* x_enc    = (const float*)d_in[0];
    const float* x_mark   = (const float*)d_in[1];
    // d_in[2], d_in[3] (x_dec / x_mark_dec) unused by the reference math
    const float* W_emb    = (const float*)d_in[4];
    const float* b_emb    = (const float*)d_in[5];
    const float* A_log_f  = (const float*)d_in[6];
    const float* Wx_f     = (const float*)d_in[7];
    const float* Wdt_f    = (const float*)d_in[8];
    const float* bdt_f    = (const float*)d_in[9];
    const float* Dskip_f  = (const float*)d_in[10];
    const float* A_log_b  = (const float*)d_in[11];
    const float* Wx_b     = (const float*)d_in[12];
    const float* Wdt_b    = (const float*)d_in[13];
    const float* bdt_b    = (const float*)d_in[14];
    const float* Dskip_b  = (const float*)d_in[15];
    const float* Wz       = (const float*)d_in[16];
    const float* bz       = (const float*)d_in[17];
    const float* Wo       = (const float*)d_in[18];
    const float* bo       = (const float*)d_in[19];
    const float* ln_g     = (const float*)d_in[20];
    const float* ln_b     = (const float*)d_in[21];
    const float* W_proj   = (const float*)d_in[22];
    const float* b_proj   = (const float*)d_in[23];
    float* out = (float*)d_out;

    // workspace layout (floats; every sub-buffer offset is a multiple of 4
    // floats -> 16B aligned for b128 loads)
    float* ws = (float*)d_ws;
    const size_t N_STATS = (size_t)B_ * N_;           // 5136
    const size_t N_BTL   = (size_t)B_ * T_ * L_;      // 2,662,400
    const size_t N_BTD   = (size_t)B_ * T_ * D_;      // 2,662,400
    const size_t N_XZ    = (size_t)B_ * T_ * 64;      //   332,800
    float* w_mean = ws;                               // B*N
    float* w_std  = w_mean + N_STATS;                 // B*N
    float* w_tok  = w_std  + N_STATS;                 // B*T*L  (reused as z/fused)
    float* w_enc  = w_tok  + N_BTL;                   // B*T*D
    float* w_xzf  = w_enc  + N_BTD;                   // B*T*64
    float* w_xzb  = w_xzf  + N_XZ;                    // B*T*64
    float* w_dt   = w_xzb  + N_XZ;                    // B*T*D  (dt_f, then dt_b, then h)
    float* w_ycat = w_dt   + N_BTD;                   // B*T*2D (reused as proj out)
    (void)ws_size; (void)in_sizes; (void)n_in; (void)out_size;

    const int Mtiles = MROWS / 16;                    // 325
    auto gemm_blocks = [](int groups) { return (groups + 7) / 8; };  // 8 waves/block

    // 1) stats
    stats_kernel<<<B_ * N_, 256, 0, stream>>>(x_enc, w_mean, w_std);

    // 2) tok
    tok_kernel<<<(int)(N_BTL / 256), 256, 0, stream>>>(x_enc, x_mark, w_mean, w_std, w_tok);

    // 3) enc = tok @ W_emb + b_emb          (N=512 -> NT=4, 8 groups)
    {
        int groups = Mtiles * (D_ / 64);
        wmma_gemm_kernel<4><<<gemm_blocks(groups), 256, 0, stream>>>(
            w_tok, L_, W_emb, L_, D_, b_emb, nullptr, 0, w_enc, D_, Mtiles, 0);
    }

    // 4) xz_f = enc @ Wx_f ; xz_b = enc @ Wx_b   (N=64 -> NT=4, 1 group)
    {
        int groups = Mtiles * (64 / 64);
        wmma_gemm_kernel<4><<<gemm_blocks(groups), 256, 0, stream>>>(
            w_enc, D_, Wx_f, D_, 64, nullptr, nullptr, 0, w_xzf, 64, Mtiles, 0);
        wmma_gemm_kernel<4><<<gemm_blocks(groups), 256, 0, stream>>>(
            w_enc, D_, Wx_b, D_, 64, nullptr, nullptr, 0, w_xzb, 64, Mtiles, 0);
    }

    // 5) dt_f = softplus(xz_f[:, :32] @ Wdt_f + bdt_f) ; forward scan
    {
        int groups = Mtiles * (D_ / 64);
        wmma_gemm_kernel<4><<<gemm_blocks(groups), 256, 0, stream>>>(
            w_xzf, 64, Wdt_f, DTR_, D_, bdt_f, nullptr, 0, w_dt, D_, Mtiles, 2);
        dim3 g(D_ / 256, B_);
        scan_kernel<<<g, 256, 0, stream>>>(w_enc, w_dt, w_xzf, A_log_f, Dskip_f, w_ycat, 0);
    }

    // 6) dt_b (reuse w_dt) ; backward scan
    {
        int groups = Mtiles * (D_ / 64);
        wmma_gemm_kernel<4><<<gemm_blocks(groups), 256, 0, stream>>>(
            w_xzb, 64, Wdt_b, DTR_, D_, bdt_b, nullptr, 0, w_dt, D_, Mtiles, 2);
        dim3 g(D_ / 256, B_);
        scan_kernel<<<g, 256, 0, stream>>>(w_enc, w_dt, w_xzb, A_log_b, Dskip_b, w_ycat, 1);
    }

    // 7) z = sigmoid(ycat @ Wz + bz)   (into w_tok, now free)
    {
        int groups = Mtiles * (D_ / 64);
        wmma_gemm_kernel<4><<<gemm_blocks(groups), 256, 0, stream>>>(
            w_ycat, 2 * D_, Wz, 2 * D_, D_, bz, nullptr, 0, w_tok, D_, Mtiles, 1);
    }

    // 8) fused = z*yf + (1-z)*yb   (in place over z)
    fuse_kernel<<<(int)(N_BTD / 256), 256, 0, stream>>>(w_ycat, w_tok);

    // 9) h = enc + fused @ Wo + bo   (into w_dt, now free)
    {
        int groups = Mtiles * (D_ / 64);
        wmma_gemm_kernel<4><<<gemm_blocks(groups), 256, 0, stream>>>(
            w_tok, D_, Wo, D_, D_, bo, w_enc, D_, w_dt, D_, Mtiles, 0);
    }

    // 10) LayerNorm in place
    ln_kernel<<<MROWS, 256, 0, stream>>>(w_dt, ln_g, ln_b);

    // 11) proj = h @ W_proj + b_proj   (into w_ycat, now free; N=96 -> NT=2, 3 groups)
    {
        int groups = Mtiles * (PRED_ / 32);
        wmma_gemm_kernel<2><<<gemm_blocks(groups), 256, 0, stream>>>(
            w_dt, D_, W_proj, D_, PRED_, b_proj, nullptr, 0, w_ycat, PRED_, Mtiles, 0);
    }

    // 12) transpose + de-normalize into d_out
    {
        size_t total = (size_t)B_ * PRED_ * N_;
        out_kernel<<<(int)((total + 255) / 256), 256, 0, stream>>>(w_ycat, w_mean, w_std, out);
    }
}